// GIN_73126113181760
// MI455X (gfx1250) — compile-verified
//
#include <hip/hip_runtime.h>

#define NNODES 100000
#define NEDGES 3200000
#define NB     512
#define CH     128
#define LDSW   132          // 128 + 4 DWORD pad per row (TDM pad_enable)
#define BN_EPS 1e-5f

typedef __attribute__((ext_vector_type(2))) float v2f;
typedef __attribute__((ext_vector_type(8))) float v8f;
typedef __attribute__((ext_vector_type(4))) unsigned int v4u;
typedef __attribute__((ext_vector_type(8))) int v8i;
typedef __attribute__((ext_vector_type(4))) int v4i;

// ---------------------------------------------------------------------------
// Edge scatter-add: agg[dst] += x[src].  One thread = (edge, 4 channels).
// Dominant roofline cost: ~1.7 GB gather + 410M f32 atomics per layer into an
// L2-resident (51 MB << 192 MB) destination.
// ---------------------------------------------------------------------------
__global__ void scatter_add_kernel(const float* __restrict__ x,
                                   const long long* __restrict__ ei,
                                   float* __restrict__ agg) {
    long long tid = (long long)blockIdx.x * blockDim.x + threadIdx.x;
    long long e = tid >> 5;
    if (e >= NEDGES) return;
    int c4 = ((int)tid & 31) * 4;
    int src = (int)ei[e];
    int dst = (int)ei[(long long)NEDGES + e];
    const float4 v = *(const float4*)(x + (long long)src * CH + c4);
    float* base = agg + (long long)dst * CH + c4;
    atomicAdd(base + 0, v.x);
    atomicAdd(base + 1, v.y);
    atomicAdd(base + 2, v.z);
    atomicAdd(base + 3, v.w);
}

// ---------------------------------------------------------------------------
// One-time: transpose all eight 128x128 weight matrices into workspace so the
// GEMM B-operand is K-contiguous (WT[n][k]).  64 KB each -- negligible.
// ---------------------------------------------------------------------------
__global__ void transpose_weights_kernel(const float* w0, const float* w1,
                                         const float* w2, const float* w3,
                                         const float* w4, const float* w5,
                                         const float* w6, const float* w7,
                                         float* __restrict__ wt) {
    const float* src;
    switch (blockIdx.x) {
        case 0: src = w0; break;  case 1: src = w1; break;
        case 2: src = w2; break;  case 3: src = w3; break;
        case 4: src = w4; break;  case 5: src = w5; break;
        case 6: src = w6; break;  default: src = w7; break;
    }
    float* dst = wt + (size_t)blockIdx.x * CH * CH;
    for (int i = threadIdx.x; i < CH * CH; i += blockDim.x) {
        int r = i >> 7;
        int c = i & (CH - 1);
        dst[c * CH + r] = src[i];   // WT[n][k] = W[k][n]
    }
}

// ---------------------------------------------------------------------------
// TDM stage: one wave DMAs the 128x128 f32 WT tile into LDS, with the TDM
// inserting 4 pad DWORDs after every 128 DWORDs (LDS row stride = 132 floats
// => conflict-free ds_load_b64 B-fragments).  D# per CDNA5 ISA §8.3/8.4.
// ---------------------------------------------------------------------------
__device__ __forceinline__ void tdm_load_weights(const float* WT, float* sW) {
    unsigned long long ga = (unsigned long long)(size_t)WT;
    unsigned int lds = (unsigned int)(size_t)sW;   // low 32 bits = LDS offset

    v4u g0;
    g0[0] = 1u;                                        // count=1, user mode
    g0[1] = lds;                                       // lds_addr [63:32]
    g0[2] = (unsigned int)(ga & 0xFFFFFFFFu);          // global_addr [95:64]
    g0[3] = (unsigned int)((ga >> 32) & 0x1FFFFFFu)    // global_addr [120:96]
          | (2u << 30);                                // type=2 ("image")

    v8i g1;
    g1[0] = (int)((2u << 16)       // data_size=2 (4 bytes), workgroup_mask=0
          | (1u << 20)             // pad_enable
          | (6u << 22)             // pad_interval: 2^(6+1)=128 DWORDs
          | (3u << 25));           // pad_amount: 3+1 = 4 DWORDs
    g1[1] = (int)(128u << 16);     // tensor_dim0[15:0]=128  (bits 79:48)
    g1[2] = (int)(128u << 16);     // tensor_dim0 hi=0, tensor_dim1[15:0]=128
    g1[3] = (int)(128u << 16);     // tensor_dim1 hi=0, tile_dim0=128
    g1[4] = 128;                   // tile_dim1=128, tile_dim2=0
    g1[5] = 128;                   // tensor_dim0_stride[31:0]=128
    g1[6] = 0;
    g1[7] = 0;

    v4i gz = {0, 0, 0, 0};         // groups 2/3 unused for 2D tensor
#if defined(__clang_major__) && (__clang_major__ >= 23)
    v8i gz8 = {0, 0, 0, 0, 0, 0, 0, 0};
    __builtin_amdgcn_tensor_load_to_lds(g0, g1, gz, gz, gz8, 0);
#else
    __builtin_amdgcn_tensor_load_to_lds(g0, g1, gz, gz, 0);
#endif
    __builtin_amdgcn_s_wait_tensorcnt(0);   // DMA complete -> LDS visible
}

// ---------------------------------------------------------------------------
// Fused GEMM (out = f(A) @ W + bias, optional post-ReLU) using fp32 WMMA.
// Block = 256 threads (8 waves); block tile = 32 rows x 128 cols.
// Wave w: rows m0 = blk*32 + (w>>2)*16, cols (w&3)*32 .. +31 (two 16x16 tiles).
// B operand comes from the padded WT stage: one ds_load_b64 per fragment.
// PRE_BN: A element a -> relu(a*scale[k] + shift[k])   (BN+ReLU fused on load)
// ---------------------------------------------------------------------------
template<bool PRE_BN, bool POST_RELU>
__global__ __launch_bounds__(256)
void gemm_bn_kernel(const float* __restrict__ A, const float* __restrict__ WT,
                    const float* __restrict__ bias,
                    const float* __restrict__ scale, const float* __restrict__ shift,
                    float* __restrict__ out) {
    extern __shared__ float sW[];   // CH * LDSW floats, filled by TDM

    if (threadIdx.x < 32) {         // wave 0 issues the async tensor DMA
        tdm_load_weights(WT, sW);
    }
    __syncthreads();

    int tid  = threadIdx.x;
    int wave = tid >> 5;
    int lane = tid & 31;
    int lh   = lane >> 4;   // half-wave: 0 or 1
    int ll   = lane & 15;
    int m0   = blockIdx.x * 32 + (wave >> 2) * 16;
    int n0   = (wave & 3) * 32;

    v8f c0 = {};
    v8f c1 = {};

    const float* arow = A + (long long)(m0 + ll) * CH;
    int n1 = n0 + ll;
    const float* brow0 = sW + n1 * LDSW;          // WT row n1 (K-contiguous)
    const float* brow1 = sW + (n1 + 16) * LDSW;   // WT row n1+16

    for (int k = 0; k < CH; k += 4) {
        int kk = k + 2 * lh;                       // K index this half-wave owns
        v2f a = *(const v2f*)(arow + kk);          // A: K=kk, kk+1
        if (PRE_BN) {
            v2f sc = *(const v2f*)(scale + kk);
            v2f sh = *(const v2f*)(shift + kk);
            a.x = fmaxf(a.x * sc.x + sh.x, 0.0f);
            a.y = fmaxf(a.y * sc.y + sh.y, 0.0f);
        }
        v2f b0 = *(const v2f*)(brow0 + kk);        // single ds_load_b64
        v2f b1 = *(const v2f*)(brow1 + kk);        // single ds_load_b64
        c0 = __builtin_amdgcn_wmma_f32_16x16x4_f32(false, a, false, b0,
                                                   (short)0, c0, false, false);
        c1 = __builtin_amdgcn_wmma_f32_16x16x4_f32(false, a, false, b1,
                                                   (short)0, c1, false, false);
    }

    float bz0 = bias[n1];
    float bz1 = bias[n1 + 16];
    int rowbase = m0 + 8 * lh;                     // C/D layout: VGPR r -> M=r(+8)
    #pragma unroll
    for (int r = 0; r < 8; ++r) {
        float v0 = c0[r] + bz0;
        float v1 = c1[r] + bz1;
        if (POST_RELU) { v0 = fmaxf(v0, 0.0f); v1 = fmaxf(v1, 0.0f); }
        float* o = out + (long long)(rowbase + r) * CH;
        o[n1]      = v0;
        o[n1 + 16] = v1;
    }
}

// ---------------------------------------------------------------------------
// BatchNorm statistics: per-channel sum / sumsq over all N rows.
// ---------------------------------------------------------------------------
__global__ void bn_stats_kernel(const float* __restrict__ h,
                                float* __restrict__ stats) {
    int c = threadIdx.x;
    float s = 0.0f, s2 = 0.0f;
    for (int row = blockIdx.x; row < NNODES; row += gridDim.x) {
        float v = h[(long long)row * CH + c];
        s += v; s2 += v * v;
    }
    atomicAdd(&stats[c], s);
    atomicAdd(&stats[CH + c], s2);
}

__global__ void bn_finalize_kernel(const float* __restrict__ stats,
                                   const float* __restrict__ gamma,
                                   const float* __restrict__ beta,
                                   float* __restrict__ scale,
                                   float* __restrict__ shift) {
    int c = threadIdx.x;
    float mean = stats[c] * (1.0f / (float)NNODES);
    float var  = stats[CH + c] * (1.0f / (float)NNODES) - mean * mean;
    float rstd = rsqrtf(var + BN_EPS);
    float sc   = rstd * gamma[c];
    scale[c] = sc;
    shift[c] = beta[c] - mean * sc;
}

// ---------------------------------------------------------------------------
// global_add_pool: g[batch[n]] += x[n]   (destination 256 KB, L2-resident)
// ---------------------------------------------------------------------------
__global__ void pool_kernel(const float* __restrict__ x,
                            const long long* __restrict__ batch,
                            float* __restrict__ g) {
    long long tid = (long long)blockIdx.x * blockDim.x + threadIdx.x;
    if (tid >= (long long)NNODES * CH) return;
    int node = (int)(tid >> 7);
    int c    = (int)(tid & (CH - 1));
    int b    = (int)batch[node];
    atomicAdd(&g[(long long)b * CH + c], x[tid]);
}

// ---------------------------------------------------------------------------
extern "C" void kernel_launch(void* const* d_in, const int* in_sizes, int n_in,
                              void* d_out, int out_size, void* d_ws, size_t ws_size,
                              hipStream_t stream) {
    const float*     x     = (const float*)d_in[0];
    const long long* ei    = (const long long*)d_in[1];
    const long long* batch = (const long long*)d_in[2];
    // d_in[3] = batch_size scalar (512, compile-time constant here)

    // Workspace layout (~103.5 MB)
    float* bufA  = (float*)d_ws;
    float* bufB  = bufA + (size_t)NNODES * CH;
    float* stats = bufB + (size_t)NNODES * CH;   // sum[128], sumsq[128]
    float* scale = stats + 2 * CH;
    float* shift = scale + CH;
    float* g     = shift + CH;                   // [512][128]
    float* g1    = g + (size_t)NB * CH;          // [512][128]
    float* wt    = g1 + (size_t)NB * CH;         // 8 x [128][128] transposed W

    const size_t lds_bytes = (size_t)CH * LDSW * sizeof(float);  // 67,584 B

    // Transpose all weight matrices up front (order: W1,W2 per layer, Wh1, Wh2)
    transpose_weights_kernel<<<8, 256, 0, stream>>>(
        (const float*)d_in[4],  (const float*)d_in[8],    // L0: W1, W2
        (const float*)d_in[10], (const float*)d_in[14],   // L1: W1, W2
        (const float*)d_in[16], (const float*)d_in[20],   // L2: W1, W2
        (const float*)d_in[22], (const float*)d_in[24],   // head: Wh1, Wh2
        wt);

    const long long sc_total = (long long)NEDGES * 32;
    const int sc_grid = (int)((sc_total + 255) / 256);

    const float* cur = x;
    float* ping = bufA;
    float* pong = bufB;

    for (int l = 0; l < 3; ++l) {
        const float* b1    = (const float*)d_in[4 + 6 * l + 1];
        const float* gamma = (const float*)d_in[4 + 6 * l + 2];
        const float* beta  = (const float*)d_in[4 + 6 * l + 3];
        const float* b2    = (const float*)d_in[4 + 6 * l + 5];
        const float* WT1   = wt + (size_t)(2 * l + 0) * CH * CH;
        const float* WT2   = wt + (size_t)(2 * l + 1) * CH * CH;

        float* agg = ping;
        float* h   = pong;

        // agg = x  (GIN eps=0: (1+eps)*x), then agg += sum_{j in N(i)} x_j
        hipMemcpyAsync(agg, cur, (size_t)NNODES * CH * sizeof(float),
                       hipMemcpyDeviceToDevice, stream);
        scatter_add_kernel<<<sc_grid, 256, 0, stream>>>(cur, ei, agg);

        // h = agg @ W1 + b1
        gemm_bn_kernel<false, false><<<NNODES / 32, 256, lds_bytes, stream>>>(
            agg, WT1, b1, nullptr, nullptr, h);

        // BN batch statistics over h
        hipMemsetAsync(stats, 0, 2 * CH * sizeof(float), stream);
        bn_stats_kernel<<<1024, CH, 0, stream>>>(h, stats);
        bn_finalize_kernel<<<1, CH, 0, stream>>>(stats, gamma, beta, scale, shift);

        // x_next = relu( relu(BN(h)) @ W2 + b2 )   (BN+ReLU fused into A load)
        gemm_bn_kernel<true, true><<<NNODES / 32, 256, lds_bytes, stream>>>(
            h, WT2, b2, scale, shift, agg);

        cur = agg;
        float* t = ping; ping = pong; pong = t;
    }

    // global_add_pool
    hipMemsetAsync(g, 0, (size_t)NB * CH * sizeof(float), stream);
    const long long pl_total = (long long)NNODES * CH;
    pool_kernel<<<(int)((pl_total + 255) / 256), 256, 0, stream>>>(cur, batch, g);

    // Head MLP: relu(g @ Wh1 + bh1) @ Wh2 + bh2
    const float* bh1 = (const float*)d_in[4 + 6 * 3 + 1];
    const float* bh2 = (const float*)d_in[4 + 6 * 3 + 3];
    const float* WTh1 = wt + (size_t)6 * CH * CH;
    const float* WTh2 = wt + (size_t)7 * CH * CH;

    gemm_bn_kernel<false, true><<<NB / 32, 256, lds_bytes, stream>>>(
        g, WTh1, bh1, nullptr, nullptr, g1);
    gemm_bn_kernel<false, false><<<NB / 32, 256, lds_bytes, stream>>>(
        g1, WTh2, bh2, nullptr, nullptr, (float*)d_out);
}